// HungarianMatcher_17514876634196
// MI455X (gfx1250) — compile-verified
//
#include <hip/hip_runtime.h>

typedef __attribute__((ext_vector_type(2))) float v2f;
typedef __attribute__((ext_vector_type(8))) float v8f;

#define BB 64
#define QQ 900
#define NC 91     // classes
#define TT 300
#define QT 16     // q rows per block
#define NTT 19    // ceil(304/16) t-tiles of 16
#define KB 23     // ceil(92/4) k-blocks for 16x16x4 wmma

__global__ __launch_bounds__(256) void matcher_cost_kernel(
    const float* __restrict__ pred_logits,   // [B,Q,91]
    const float* __restrict__ pred_boxes,    // [B,Q,4]
    const float* __restrict__ tgt_boxes,     // [B,T,4]
    const int*   __restrict__ tgt_labels,    // [B,T]
    float* __restrict__ out)                 // [B,Q,T]
{
    __shared__ float probs[QT][92];   // normalized softmax probs (col 91 == 0 pad)
    __shared__ float tgt[304][9];     // cx,cy,w,h, X0,Y0,X1,Y1, area (double-converted)
    __shared__ int   tlab[304];
    __shared__ float pq[QT][9];       // same 9 fields for pred boxes

    const int tid = threadIdx.x;
    const int b   = blockIdx.y;
    const int q0  = blockIdx.x * QT;

    // ---------- softmax probabilities for 16 q-rows (16 threads per row) ----------
    {
        const int row = tid >> 4;          // 0..15
        const int sub = tid & 15;
        const int q   = q0 + row;
        const float* lg = pred_logits + (size_t)(b * QQ + q) * NC;
        float v[6];
        float mx = -INFINITY;
        #pragma unroll
        for (int k = 0; k < 6; ++k) {
            int c = sub + 16 * k;
            float x = -INFINITY;
            if (c < 92) {
                if (q < QQ) x = (c < NC) ? lg[c] : -INFINITY;  // pad col -> prob 0
                else        x = 0.0f;                          // dead row, masked later
            }
            v[k] = x;
            mx = fmaxf(mx, x);
        }
        #pragma unroll
        for (int m = 8; m >= 1; m >>= 1) mx = fmaxf(mx, __shfl_xor(mx, m, 16));
        float e[6];
        float s = 0.0f;
        #pragma unroll
        for (int k = 0; k < 6; ++k) {
            int c = sub + 16 * k;
            float ev = 0.0f;
            if (c < 92) ev = __expf(v[k] - mx);   // exp(-inf)=0 for pad col
            e[k] = ev;
            s += ev;
        }
        #pragma unroll
        for (int m = 8; m >= 1; m >>= 1) s += __shfl_xor(s, m, 16);
        const float inv = 1.0f / s;
        #pragma unroll
        for (int k = 0; k < 6; ++k) {
            int c = sub + 16 * k;
            if (c < 92) probs[row][c] = e[k] * inv;
        }
    }

    // ---------- stage targets: raw box, double-converted xyxy, area, label ----------
    for (int t = tid; t < 304; t += 256) {
        float cx = 0.f, cy = 0.f, w = 0.f, h = 0.f;
        int lab = -1;
        if (t < TT) {
            const float* tb = tgt_boxes + (size_t)(b * TT + t) * 4;
            cx = tb[0]; cy = tb[1]; w = tb[2]; h = tb[3];
            lab = tgt_labels[b * TT + t];
        }
        float x0 = cx - 0.5f * w, y0 = cy - 0.5f * h;
        float x1 = cx + 0.5f * w, y1 = cy + 0.5f * h;
        float X0 = x0 - 0.5f * x1, Y0 = y0 - 0.5f * y1;     // faithful double conversion
        float X1 = x0 + 0.5f * x1, Y1 = y0 + 0.5f * y1;
        tgt[t][0] = cx; tgt[t][1] = cy; tgt[t][2] = w;  tgt[t][3] = h;
        tgt[t][4] = X0; tgt[t][5] = Y0; tgt[t][6] = X1; tgt[t][7] = Y1;
        tgt[t][8] = (X1 - X0) * (Y1 - Y0);
        tlab[t] = lab;
    }

    // ---------- stage pred box data for the 16 q-rows ----------
    if (tid < QT) {
        int q = q0 + tid;
        float cx = 0.f, cy = 0.f, w = 0.f, h = 0.f;
        if (q < QQ) {
            const float* pb = pred_boxes + (size_t)(b * QQ + q) * 4;
            cx = pb[0]; cy = pb[1]; w = pb[2]; h = pb[3];
        }
        float x0 = cx - 0.5f * w, y0 = cy - 0.5f * h;
        float x1 = cx + 0.5f * w, y1 = cy + 0.5f * h;
        float X0 = x0 - 0.5f * x1, Y0 = y0 - 0.5f * y1;
        float X1 = x0 + 0.5f * x1, Y1 = y0 + 0.5f * y1;
        pq[tid][0] = cx; pq[tid][1] = cy; pq[tid][2] = w;  pq[tid][3] = h;
        pq[tid][4] = X0; pq[tid][5] = Y0; pq[tid][6] = X1; pq[tid][7] = Y1;
        pq[tid][8] = (X1 - X0) * (Y1 - Y0);
    }
    __syncthreads();

    // ---------- main: one 16x16 output tile per wave per iteration ----------
    const int wave = tid >> 5;
    const int lane = tid & 31;
    const int n    = lane & 15;          // output column (t) / A row (m) within tile
    const int half = lane >> 4;          // lane half selects K offset
    const int koff = half << 1;

    const size_t outbase = (size_t)b * QQ * TT;

    for (int tt = wave; tt < NTT; tt += 8) {
        const int t   = tt * 16 + n;     // <= 303, LDS is padded to 304
        const int lab = tlab[t];

        const float tcx = tgt[t][0], tcy = tgt[t][1], tw = tgt[t][2], th = tgt[t][3];
        const float tX0 = tgt[t][4], tY0 = tgt[t][5], tX1 = tgt[t][6], tY1 = tgt[t][7];
        const float tA  = tgt[t][8];

        v8f acc = {0.f, 0.f, 0.f, 0.f, 0.f, 0.f, 0.f, 0.f};

        // cost_class as P[16x92] x onehot(labels)[92x16] — exact (single nonzero per column)
        #pragma unroll
        for (int kb = 0; kb < KB; ++kb) {
            const int c0 = kb * 4 + koff;
            v2f a;                              // A: 16x4 fp32 fragment
            a.x = probs[n][c0];
            a.y = probs[n][c0 + 1];
            v2f bf;                             // B: 4x16 one-hot fragment
            bf.x = (lab == c0)     ? 1.0f : 0.0f;
            bf.y = (lab == c0 + 1) ? 1.0f : 0.0f;
            acc = __builtin_amdgcn_wmma_f32_16x16x4_f32(
                false, a, false, bf, (short)0, acc, false, false);
        }

        // fuse bbox-L1 and GIoU with the accumulated class probability
        #pragma unroll
        for (int r = 0; r < 8; ++r) {
            const int m = r + (half << 3);     // C/D layout: M = r + 8*(lane>>4)
            const int q = q0 + m;
            const float S = acc[r];            // softmax prob at target label

            const float pcx = pq[m][0], pcy = pq[m][1], pw = pq[m][2], ph = pq[m][3];
            const float pX0 = pq[m][4], pY0 = pq[m][5], pX1 = pq[m][6], pY1 = pq[m][7];
            const float pA  = pq[m][8];

            float l1 = fabsf(pcx - tcx) + fabsf(pcy - tcy)
                     + fabsf(pw  - tw)  + fabsf(ph  - th);

            float lx = fmaxf(pX0, tX0), ly = fmaxf(pY0, tY0);
            float rx = fminf(pX1, tX1), ry = fminf(pY1, tY1);
            float iw = fmaxf(rx - lx, 0.f), ih = fmaxf(ry - ly, 0.f);
            float inter = iw * ih;
            float uni   = pA + tA - inter;
            float iou   = inter / uni;

            float ex = fminf(pX0, tX0), ey = fminf(pY0, tY0);
            float fx = fmaxf(pX1, tX1), fy = fmaxf(pY1, tY1);
            float cw = fmaxf(fx - ex, 0.f), ch = fmaxf(fy - ey, 0.f);
            float ac = cw * ch;
            float giou = iou - (ac - uni) / ac;

            float cost = 5.0f * l1 - S - 2.0f * giou;

            if (q < QQ && t < TT)
                out[outbase + (size_t)q * TT + t] = cost;
        }
    }
}

extern "C" void kernel_launch(void* const* d_in, const int* in_sizes, int n_in,
                              void* d_out, int out_size, void* d_ws, size_t ws_size,
                              hipStream_t stream) {
    (void)in_sizes; (void)n_in; (void)d_ws; (void)ws_size; (void)out_size;
    const float* pred_logits = (const float*)d_in[0];
    const float* pred_boxes  = (const float*)d_in[1];
    const float* tgt_boxes   = (const float*)d_in[2];
    const int*   tgt_labels  = (const int*)d_in[3];
    float* out = (float*)d_out;

    dim3 grid((QQ + QT - 1) / QT, BB);   // 57 x 64 blocks
    matcher_cost_kernel<<<grid, 256, 0, stream>>>(
        pred_logits, pred_boxes, tgt_boxes, tgt_labels, out);
}